// CDSSMBlock_33028298506325
// MI455X (gfx1250) — compile-verified
//
#include <hip/hip_runtime.h>
#include <hip/hip_bf16.h>
#include <math.h>

typedef __attribute__((ext_vector_type(2))) float v2f;
typedef __attribute__((ext_vector_type(8))) float v8f;

#define Bn   4
#define Ln   2048
#define DMn  1024
#define DIn  2048
#define Hn   16
#define HDn  128
#define PROJN 6176          // 3*DI + 2*H
#define BLn  (Bn*Ln)        // 8192

__device__ __forceinline__ float sigm(float x)     { return 1.0f / (1.0f + expf(-x)); }
__device__ __forceinline__ float softplus(float x) { return (x > 20.0f) ? x : log1pf(expf(x)); }
__device__ __forceinline__ float wave_sum(float v) {
  #pragma unroll
  for (int m = 16; m >= 1; m >>= 1) v += __shfl_xor(v, m, 32);
  return v;
}

// ---------------------------------------------------------------- RMS norm
__global__ __launch_bounds__(256) void rms_kernel(const float* __restrict__ x,
                                                  const float* __restrict__ w,
                                                  float* __restrict__ xn) {
  int row = blockIdx.x;
  const float* xr = x + (size_t)row * DMn;
  float s = 0.0f;
  for (int i = threadIdx.x; i < DMn; i += 256) { float v = xr[i]; s += v * v; }
  __shared__ float red[256];
  red[threadIdx.x] = s; __syncthreads();
  for (int off = 128; off > 0; off >>= 1) {
    if (threadIdx.x < off) red[threadIdx.x] += red[threadIdx.x + off];
    __syncthreads();
  }
  float inv = rsqrtf(red[0] / (float)DMn + 1e-6f);
  for (int i = threadIdx.x; i < DMn; i += 256)
    xn[(size_t)row * DMn + i] = xr[i] * inv * w[i];
}

// -------------------------------------------- fp32 WMMA GEMM, 64x128 block tile
// C[M,N] = A[M,K]*B[K,N] (+bias[N]) (+resid[M,N]); row-major. 8 waves, each wave
// owns a 16x64 strip (4 accumulators). Requires M%64==0, K%16==0 (true for all
// calls); only N may be clipped. Full tiles: double-buffered async b128 staging.
__global__ __launch_bounds__(256) void gemm_wmma_f32(
    const float* __restrict__ A, const float* __restrict__ Bm,
    const float* __restrict__ bias, const float* __restrict__ resid,
    float* __restrict__ C, int M, int N, int K, int lda, int ldb, int ldc) {
  __shared__ __align__(16) float sA[2][64][20];    // row stride 80B
  __shared__ __align__(16) float sB[2][16][136];   // row stride 544B
  const int tid  = threadIdx.x;
  const int lane = tid & 31, wid = tid >> 5;
  const int wrow = wid >> 1, wcol = wid & 1;       // 4x2 waves -> 64x128
  const int l16  = lane & 15, hi = lane >> 4;
  const int bm = blockIdx.y * 64, bn = blockIdx.x * 128;
  v8f acc[4] = {{}, {}, {}, {}};
  const bool full = (bn + 128 <= N);

  const int am = tid >> 2, ak = (tid & 3) << 2;    // A: 64 rows x 4 chunks of 4
  const float* Ap = A + (size_t)(bm + am) * lda + ak;
  unsigned la[2] = { (unsigned)(uintptr_t)&sA[0][am][ak], (unsigned)(uintptr_t)&sA[1][am][ak] };

  if (full) {
    // B: 16 rows x 32 chunks of 4 = 512 chunks, 2 per thread
    const int bk0 = tid >> 5,        bn0 = (tid & 31) << 2;
    const int bk1 = (tid + 256) >> 5, bn1 = ((tid + 256) & 31) << 2;
    const float* Bp0 = Bm + (size_t)bk0 * ldb + (bn + bn0);
    const float* Bp1 = Bm + (size_t)bk1 * ldb + (bn + bn1);
    unsigned lb0[2] = { (unsigned)(uintptr_t)&sB[0][bk0][bn0], (unsigned)(uintptr_t)&sB[1][bk0][bn0] };
    unsigned lb1[2] = { (unsigned)(uintptr_t)&sB[0][bk1][bn1], (unsigned)(uintptr_t)&sB[1][bk1][bn1] };
    {   // prologue: tile 0 -> buffer 0
      unsigned long long ga  = (unsigned long long)(uintptr_t)Ap;
      unsigned long long gb0 = (unsigned long long)(uintptr_t)Bp0;
      unsigned long long gb1 = (unsigned long long)(uintptr_t)Bp1;
      asm volatile("global_load_async_to_lds_b128 %0, %1, off" :: "v"(la[0]),  "v"(ga)  : "memory");
      asm volatile("global_load_async_to_lds_b128 %0, %1, off" :: "v"(lb0[0]), "v"(gb0) : "memory");
      asm volatile("global_load_async_to_lds_b128 %0, %1, off" :: "v"(lb1[0]), "v"(gb1) : "memory");
    }
    int cur = 0;
    for (int k0 = 0; k0 < K; k0 += 16) {
      if (k0 + 16 < K) {   // issue next tile, then wait for current (in-order)
        unsigned long long ga  = (unsigned long long)(uintptr_t)(Ap + (k0 + 16));
        unsigned long long gb0 = (unsigned long long)(uintptr_t)(Bp0 + (size_t)(k0 + 16) * ldb);
        unsigned long long gb1 = (unsigned long long)(uintptr_t)(Bp1 + (size_t)(k0 + 16) * ldb);
        asm volatile("global_load_async_to_lds_b128 %0, %1, off" :: "v"(la[cur ^ 1]),  "v"(ga)  : "memory");
        asm volatile("global_load_async_to_lds_b128 %0, %1, off" :: "v"(lb0[cur ^ 1]), "v"(gb0) : "memory");
        asm volatile("global_load_async_to_lds_b128 %0, %1, off" :: "v"(lb1[cur ^ 1]), "v"(gb1) : "memory");
        asm volatile("s_wait_asynccnt 0x3" ::: "memory");
      } else {
        asm volatile("s_wait_asynccnt 0x0" ::: "memory");
      }
      __syncthreads();
      const float (*cA)[20]  = sA[cur];
      const float (*cB)[136] = sB[cur];
      #pragma unroll
      for (int kk = 0; kk < 16; kk += 4) {
        int ksel = kk + 2 * hi;                 // 32-bit lane layout: V0={K0|K2}, V1={K1|K3}
        v2f a = *(const v2f*)&cA[wrow * 16 + l16][ksel];
        #pragma unroll
        for (int s = 0; s < 4; ++s) {
          v2f b;
          b.x = cB[ksel][wcol * 64 + s * 16 + l16];
          b.y = cB[ksel + 1][wcol * 64 + s * 16 + l16];
          acc[s] = __builtin_amdgcn_wmma_f32_16x16x4_f32(false, a, false, b, (short)0, acc[s], false, false);
        }
      }
      __syncthreads();
      cur ^= 1;
    }
  } else {
    // N-edge tiles: A still via async b128 (never clipped); B guarded scalar
    for (int k0 = 0; k0 < K; k0 += 16) {
      {
        unsigned long long ga = (unsigned long long)(uintptr_t)(Ap + k0);
        asm volatile("global_load_async_to_lds_b128 %0, %1, off" :: "v"(la[0]), "v"(ga) : "memory");
      }
      #pragma unroll
      for (int r = 0; r < 8; ++r) {            // 16x128 = 2048 elems, 8/thread
        int q = tid + r * 256;
        int krow = q >> 7, n = q & 127;
        int gn = bn + n;
        sB[0][krow][n] = (gn < N) ? Bm[(size_t)(k0 + krow) * ldb + gn] : 0.0f;
      }
      asm volatile("s_wait_asynccnt 0x0" ::: "memory");
      __syncthreads();
      #pragma unroll
      for (int kk = 0; kk < 16; kk += 4) {
        int ksel = kk + 2 * hi;
        v2f a = *(const v2f*)&sA[0][wrow * 16 + l16][ksel];
        #pragma unroll
        for (int s = 0; s < 4; ++s) {
          v2f b;
          b.x = sB[0][ksel][wcol * 64 + s * 16 + l16];
          b.y = sB[0][ksel + 1][wcol * 64 + s * 16 + l16];
          acc[s] = __builtin_amdgcn_wmma_f32_16x16x4_f32(false, a, false, b, (short)0, acc[s], false, false);
        }
      }
      __syncthreads();
    }
  }
  // epilogue; C/D layout: VGPR i -> M = i + 8*hi, N = l16
  if (full) {
    #pragma unroll
    for (int s = 0; s < 4; ++s) {
      int gn = bn + wcol * 64 + s * 16 + l16;
      float badd = bias ? bias[gn] : 0.0f;
      #pragma unroll
      for (int i = 0; i < 8; ++i) {
        int gm = bm + wrow * 16 + hi * 8 + i;
        float v = acc[s][i] + badd;
        if (resid) v += resid[(size_t)gm * ldc + gn];
        C[(size_t)gm * ldc + gn] = v;
      }
    }
  } else {
    #pragma unroll
    for (int s = 0; s < 4; ++s) {
      int gn = bn + wcol * 64 + s * 16 + l16;
      if (gn < N) {
        float badd = bias ? bias[gn] : 0.0f;
        #pragma unroll
        for (int i = 0; i < 8; ++i) {
          int gm = bm + wrow * 16 + hi * 8 + i;
          float v = acc[s][i] + badd;
          if (resid) v += resid[(size_t)gm * ldc + gn];
          C[(size_t)gm * ldc + gn] = v;
        }
      }
    }
  }
}

// ---------------------------------------------------------- depthwise conv+SiLU
__global__ __launch_bounds__(256) void conv_silu_kernel(
    const float* __restrict__ proj, const float* __restrict__ cw,
    const float* __restrict__ cb, float* __restrict__ xconv) {
  int idx = blockIdx.x * 256 + threadIdx.x;     // over B*L*DI
  int c  = idx & (DIn - 1);
  int bl = idx >> 11;                            // DIn = 2048 = 2^11
  int l  = bl & (Ln - 1);
  float y = cb[c];
  #pragma unroll
  for (int j = 0; j < 4; ++j) {
    int ls = l - 3 + j;
    float xv = (ls >= 0) ? proj[(size_t)(bl - l + ls) * PROJN + DIn + c] : 0.0f;
    y += xv * cw[c * 4 + j];
  }
  xconv[(size_t)idx] = y * sigm(y);
}

// ------------------------------------- gate math + K/Q head-norm (wave/(b,l,h))
__global__ __launch_bounds__(256) void gate_kernel(
    const float* __restrict__ gates, const float* __restrict__ proj,
    const float* __restrict__ log_dt, float* __restrict__ qbuf,
    float* __restrict__ knorm, float* __restrict__ vg, float4* __restrict__ scal) {
  int lane = threadIdx.x & 31;
  int gw = blockIdx.x * 8 + (threadIdx.x >> 5);  // one wave per (b,l,h)
  int h  = gw & (Hn - 1);
  int bl = gw >> 4;
  const float* g = gates + (size_t)bl * 112;
  float a_raw = g[0 * Hn + h], o_raw = g[1 * Hn + h];
  float selB  = g[2 * Hn + h], selC  = g[3 * Hn + h];
  float s_dt  = g[4 * Hn + h], b_raw = g[5 * Hn + h], r_raw = g[6 * Hn + h];
  float freq  = expf(-((float)h / (float)Hn) * logf(10000.0f));
  float dt    = softplus(s_dt) * expf(log_dt[h]) + 1e-4f;
  float alpha = softplus(a_raw);
  float omega = freq * (1.0f + tanhf(o_raw));
  float ah = 0.5f * dt * alpha, wh = 0.5f * dt * omega;
  float det = (1.0f + ah) * (1.0f + ah) + wh * wh;
  float c = (1.0f - ah * ah - wh * wh) / det;
  float s = 2.0f * wh / det;
  float mag = sqrtf(c * c + s * s);
  float theta = atan2f(s, c);
  float r = sigm(r_raw);
  float mag2 = expf(8.0f * r * logf(fminf(fmaxf(mag, 1e-8f), 1.0f)));
  float cbar = mag2 * cosf(theta), sbar = mag2 * sinf(theta);
  float vps = sigm(selB) * sqrtf(fminf(fmaxf(1.0f - mag2 * mag2, 0.0f), 1.0f) + 1e-6f);
  // K head-norm: proj cols [2*DI, 3*DI)
  const float* kp = proj + (size_t)bl * PROJN + 2 * DIn + h * HDn + lane * 4;
  float4 kv = *(const float4*)kp;
  float ss = wave_sum(kv.x * kv.x + kv.y * kv.y + kv.z * kv.z + kv.w * kv.w);
  float kinv = 1.0f / (sqrtf(ss) + 1e-6f);
  float4 ko = make_float4(kv.x * kinv, kv.y * kinv, kv.z * kinv, kv.w * kinv);
  *(float4*)(knorm + ((size_t)bl * Hn + h) * HDn + lane * 4) = ko;
  // Q head-norm in place
  float* qp = qbuf + (size_t)bl * DIn + h * HDn + lane * 4;
  float4 qv = *(const float4*)qp;
  float qs = wave_sum(qv.x * qv.x + qv.y * qv.y + qv.z * qv.z + qv.w * qv.w);
  float qinv = 1.0f / (sqrtf(qs) + 1e-6f);
  *(float4*)qp = make_float4(qv.x * qinv, qv.y * qinv, qv.z * qinv, qv.w * qinv);
  if (lane == 0) {
    float v0 = proj[(size_t)bl * PROJN + 3 * DIn + h * 2 + 0];
    float v1 = proj[(size_t)bl * PROJN + 3 * DIn + h * 2 + 1];
    vg[((size_t)bl * Hn + h) * 2 + 0] = v0 * vps;
    vg[((size_t)bl * Hn + h) * 2 + 1] = v1 * vps;
    scal[(size_t)bl * Hn + h] = make_float4(cbar, sbar, sigm(b_raw), sigm(selC));
  }
}

// ---------------------------------------------- sequential scan: 1 wave/(b,h)
__global__ __launch_bounds__(32) void scan_kernel(
    const float* __restrict__ knorm, const float* __restrict__ qbuf,
    const float* __restrict__ vg, const float4* __restrict__ scal,
    float* __restrict__ retr) {
  int lane = threadIdx.x;
  int b = blockIdx.x >> 4, h = blockIdx.x & (Hn - 1);
  float S0[4] = {0, 0, 0, 0}, S1[4] = {0, 0, 0, 0};
  for (int t = 0; t < Ln; ++t) {
    size_t bl  = (size_t)b * Ln + t;
    size_t blh = bl * Hn + h;
    float4 sc = scal[blh];                                 // cbar,sbar,beta,selC
    float  vx = vg[blh * 2 + 0], vy = vg[blh * 2 + 1];
    float4 kk = *(const float4*)(knorm + blh * HDn + lane * 4);
    float4 qq = *(const float4*)(qbuf + bl * DIn + h * HDn + lane * 4);
    float k[4] = {kk.x, kk.y, kk.z, kk.w};
    float q[4] = {qq.x, qq.y, qq.z, qq.w};
    float p0 = 0.0f, p1 = 0.0f;
    #pragma unroll
    for (int j = 0; j < 4; ++j) {
      float n0 =  sc.x * S0[j] + sc.y * S1[j];
      float n1 = -sc.y * S0[j] + sc.x * S1[j];
      S0[j] = n0; S1[j] = n1;
      p0 += n0 * k[j]; p1 += n1 * k[j];
    }
    p0 = wave_sum(p0); p1 = wave_sum(p1);
    float e0 = sc.z * (vx - p0), e1 = sc.z * (vy - p1);
    float r0 = 0.0f, r1 = 0.0f;
    #pragma unroll
    for (int j = 0; j < 4; ++j) {
      S0[j] += e0 * k[j]; S1[j] += e1 * k[j];
      r0 += S0[j] * q[j]; r1 += S1[j] * q[j];
    }
    r0 = wave_sum(r0); r1 = wave_sum(r1);
    if (lane == 0) {
      retr[bl * (2 * Hn) + h * 2 + 0] = r0 * sc.w;
      retr[bl * (2 * Hn) + h * 2 + 1] = r1 * sc.w;
    }
  }
}

// -------------------------------------------- GroupNorm stats per (b, group)
__global__ __launch_bounds__(256) void gnstat_kernel(const float* __restrict__ y,
                                                     float* __restrict__ stats) {
  int b = blockIdx.x >> 4, g = blockIdx.x & 15;
  float s = 0.0f, sq = 0.0f;
  for (int i = threadIdx.x; i < Ln * HDn; i += 256) {
    int l = i >> 7, c = g * HDn + (i & 127);
    float v = y[((size_t)b * Ln + l) * DIn + c];
    s += v; sq += v * v;
  }
  __shared__ float r0[256], r1[256];
  r0[threadIdx.x] = s; r1[threadIdx.x] = sq; __syncthreads();
  for (int off = 128; off > 0; off >>= 1) {
    if (threadIdx.x < off) { r0[threadIdx.x] += r0[threadIdx.x + off];
                             r1[threadIdx.x] += r1[threadIdx.x + off]; }
    __syncthreads();
  }
  if (threadIdx.x == 0) {
    float n = (float)(Ln * HDn);
    float mean = r0[0] / n;
    stats[blockIdx.x * 2 + 0] = mean;
    stats[blockIdx.x * 2 + 1] = r1[0] / n - mean * mean;
  }
}

// ----------------------------- GN apply + SiLU(z) gate + D*x_conv, in place y
__global__ __launch_bounds__(256) void fuse_kernel(
    float* __restrict__ y, const float* __restrict__ stats,
    const float* __restrict__ gnw, const float* __restrict__ gnb,
    const float* __restrict__ proj, const float* __restrict__ xconv,
    const float* __restrict__ D) {
  int idx = blockIdx.x * 256 + threadIdx.x;     // over B*L*DI
  int c  = idx & (DIn - 1);
  int bl = idx >> 11;
  int b  = bl >> 11;                             // Ln = 2048 = 2^11
  int g  = c >> 7;
  float mean = stats[(b * 16 + g) * 2 + 0];
  float var  = stats[(b * 16 + g) * 2 + 1];
  float yn = (y[(size_t)idx] - mean) * rsqrtf(var + 1e-5f) * gnw[c] + gnb[c];
  float z = proj[(size_t)bl * PROJN + c];
  y[(size_t)idx] = yn * (z * sigm(z)) + D[c] * xconv[(size_t)idx];
}

// ---------------------------------------------------------------------------
extern "C" void kernel_launch(void* const* d_in, const int* in_sizes, int n_in,
                              void* d_out, int out_size, void* d_ws, size_t ws_size,
                              hipStream_t stream) {
  const float* x        = (const float*)d_in[0];
  const float* rms_w    = (const float*)d_in[1];
  const float* in_w     = (const float*)d_in[2];
  const float* in_b     = (const float*)d_in[3];
  const float* conv_w   = (const float*)d_in[4];
  const float* conv_b   = (const float*)d_in[5];
  const float* gate_w   = (const float*)d_in[6];
  const float* gate_b   = (const float*)d_in[7];
  const float* log_dt   = (const float*)d_in[8];
  const float* Q_w      = (const float*)d_in[9];
  const float* ro_w     = (const float*)d_in[10];
  const float* out_w    = (const float*)d_in[11];
  const float* gn_w     = (const float*)d_in[12];
  const float* gn_b     = (const float*)d_in[13];
  const float* D_p      = (const float*)d_in[14];
  float* out = (float*)d_out;

  char* ws = (char*)d_ws;
  size_t off = 0;
  auto carve = [&](size_t bytes) { char* p = ws + off; off += (bytes + 255) & ~(size_t)255; return p; };
  float*  xn    = (float*)carve((size_t)BLn * DMn * 4);
  float*  proj  = (float*)carve((size_t)BLn * PROJN * 4);
  float*  xconv = (float*)carve((size_t)BLn * DIn * 4);
  float*  gates = (float*)carve((size_t)BLn * 112 * 4);
  float*  qbuf  = (float*)carve((size_t)BLn * DIn * 4);
  float*  knorm = (float*)carve((size_t)BLn * DIn * 4);
  float*  vg    = (float*)carve((size_t)BLn * Hn * 2 * 4);
  float4* scal  = (float4*)carve((size_t)BLn * Hn * 16);
  float*  retr  = (float*)carve((size_t)BLn * 2 * Hn * 4);
  float*  ybuf  = (float*)carve((size_t)BLn * DIn * 4);
  float*  stats = (float*)carve(Bn * 16 * 2 * 4);

  // 1. RMS norm
  rms_kernel<<<BLn, 256, 0, stream>>>(x, rms_w, xn);
  // 2. proj = xn @ in_proj_w + b   (8192 x 6176, K=1024)
  gemm_wmma_f32<<<dim3((PROJN + 127) / 128, BLn / 64), 256, 0, stream>>>(
      xn, in_w, in_b, nullptr, proj, BLn, PROJN, DMn, DMn, PROJN, PROJN);
  // 3. depthwise conv + SiLU on x_gate slice
  conv_silu_kernel<<<(BLn * DIn) / 256, 256, 0, stream>>>(proj, conv_w, conv_b, xconv);
  // 4. gates = x_conv @ gate_w + gate_b   (N=112)
  gemm_wmma_f32<<<dim3(1, BLn / 64), 256, 0, stream>>>(
      xconv, gate_w, gate_b, nullptr, gates, BLn, 112, DIn, DIn, 112, 112);
  // 5. Q = x_conv @ Q_w
  gemm_wmma_f32<<<dim3(DIn / 128, BLn / 64), 256, 0, stream>>>(
      xconv, Q_w, nullptr, nullptr, qbuf, BLn, DIn, DIn, DIn, DIn, DIn);
  // 6. gate scalar math + K/Q head norms + Vg
  gate_kernel<<<(BLn * Hn) / 8, 256, 0, stream>>>(gates, proj, log_dt, qbuf, knorm, vg, scal);
  // 7. sequential scan, one wave per (b,h)
  scan_kernel<<<Bn * Hn, 32, 0, stream>>>(knorm, qbuf, vg, scal, retr);
  // 8. y = retrieved @ readout_w   (K=32)
  gemm_wmma_f32<<<dim3(DIn / 128, BLn / 64), 256, 0, stream>>>(
      retr, ro_w, nullptr, nullptr, ybuf, BLn, DIn, 2 * Hn, 2 * Hn, DIn, DIn);
  // 9. GroupNorm stats over (L, group-channels) per (b,g)
  gnstat_kernel<<<Bn * 16, 256, 0, stream>>>(ybuf, stats);
  // 10. GN apply + SiLU(z) gate + D*x_conv (in place)
  fuse_kernel<<<(BLn * DIn) / 256, 256, 0, stream>>>(ybuf, stats, gn_w, gn_b, proj, xconv, D_p);
  // 11. out = x + ybuf @ out_w
  gemm_wmma_f32<<<dim3(DMn / 128, BLn / 64), 256, 0, stream>>>(
      ybuf, out_w, nullptr, x, out, BLn, DMn, DIn, DIn, DMn, DMn);
  (void)in_sizes; (void)n_in; (void)out_size; (void)ws_size; (void)off;
}